// HeteroGNN_70935679860741
// MI455X (gfx1250) — compile-verified
//
#include <hip/hip_runtime.h>

typedef float v2f __attribute__((ext_vector_type(2)));
typedef float v8f __attribute__((ext_vector_type(8)));

// ---------------------------------------------------------------------------
// K1: zero scratch (counts, per-source weights, reduction vectors)
// ---------------------------------------------------------------------------
__global__ void zero_ws_kernel(float* __restrict__ p, int n) {
    int i = blockIdx.x * blockDim.x + threadIdx.x;
    if (i < n) p[i] = 0.0f;
}

// ---------------------------------------------------------------------------
// K2: in-degree counts: cnt[dst[e]] += 1
// ---------------------------------------------------------------------------
__global__ void count_deg_kernel(const int* __restrict__ dst, int E,
                                 unsigned* __restrict__ cnt) {
    int i = blockIdx.x * blockDim.x + threadIdx.x;
    if (i < E) atomicAdd(&cnt[dst[i]], 1u);
}

// ---------------------------------------------------------------------------
// K3: cnt -> 1/max(cnt,1), stored in place as float
// ---------------------------------------------------------------------------
__global__ void invert_cnt_kernel(unsigned* __restrict__ cnt, int n) {
    int i = blockIdx.x * blockDim.x + threadIdx.x;
    if (i < n) {
        unsigned c = cnt[i];
        float inv = 1.0f / (float)(c > 0u ? c : 1u);
        ((float*)cnt)[i] = inv;
    }
}

// ---------------------------------------------------------------------------
// K4: per-source weight accumulation: a[src[e]] += inv[dst[e]]
// ---------------------------------------------------------------------------
__global__ void scatter_w_kernel(const int* __restrict__ src,
                                 const int* __restrict__ dst, int E,
                                 const float* __restrict__ inv,
                                 float* __restrict__ acc) {
    int i = blockIdx.x * blockDim.x + threadIdx.x;
    if (i < E) atomicAdd(&acc[src[i]], inv[dst[i]]);
}

// ---------------------------------------------------------------------------
// K5: single streaming pass over x producing three 128-vectors:
//     vecs[0:128)   = sum_j a_pos[j] * x[j,:]
//     vecs[128:256) = sum_j a_neg[j] * x[j,:]
//     vecs[256:384) = sum_j x[j,:]
// blockDim must be 128 (one thread per feature column)
// ---------------------------------------------------------------------------
__global__ __launch_bounds__(128)
void reduce_x_kernel(const float* __restrict__ x,
                     const float* __restrict__ a_pos,
                     const float* __restrict__ a_neg,
                     int N, float* __restrict__ vecs) {
    const int col = threadIdx.x;                     // 0..127
    float vp = 0.0f, vn = 0.0f, sx = 0.0f;
    for (int j = blockIdx.x; j < N; j += gridDim.x) {
        float val = x[(size_t)j * 128 + col];        // coalesced row read
        float ap  = a_pos[j];                        // wave-broadcast scalar
        float an  = a_neg[j];
        vp += ap * val;
        vn += an * val;
        sx += val;
    }
    atomicAdd(&vecs[col],       vp);
    atomicAdd(&vecs[128 + col], vn);
    atomicAdd(&vecs[256 + col], sx);
}

// ---------------------------------------------------------------------------
// K6: final GEMV out[1,128] = u1@Wl_pos + u2@(Wr_pos+Wr_neg) + u3@Wl_neg + b
// via V_WMMA_F32_16X16X4_F32. One block of 8 waves; wave w owns columns
// [16w, 16w+16). A-tile: only row M=0 populated (u vector, 1/N folded in).
// A layout (16x4 f32): v0 = K=0 (lanes 0-15) / K=2 (lanes 16-31); v1 = K=1/K=3.
// B layout (4x16 f32): lanes 0-15 hold K=0..1, lanes 16-31 hold K=2..3.
// D row M=0 = c[0], lanes 0-15.
// ---------------------------------------------------------------------------
__global__ __launch_bounds__(256)
void gemv_wmma_kernel(const float* __restrict__ vecs,
                      const float* __restrict__ Wlp, const float* __restrict__ Wrp,
                      const float* __restrict__ Wln, const float* __restrict__ Wrn,
                      const float* __restrict__ blp, const float* __restrict__ bln,
                      float* __restrict__ out, float invN) {
    const int lane = threadIdx.x & 31;
    const int wave = threadIdx.x >> 5;       // 0..7 -> column tile
    const int l16  = lane & 15;
    const int half = lane >> 4;              // 0: K even pair base, 1: K+2
    const int col  = wave * 16 + l16;        // output column this lane covers in B/D
    const float rowmask = (l16 == 0) ? invN : 0.0f;   // only M=0 row of A is live

    v8f c = {0.0f, 0.0f, 0.0f, 0.0f, 0.0f, 0.0f, 0.0f, 0.0f};

    const float* u1 = vecs;          // sum a_pos * x   -> Wl_pos
    const float* u2 = vecs + 256;    // sum x           -> Wr_pos + Wr_neg
    const float* u3 = vecs + 128;    // sum a_neg * x   -> Wl_neg

    // Pass 1: u1 @ Wl_pos
    for (int kb = 0; kb < 128; kb += 4) {
        const int k0 = kb + 2 * half, k1 = k0 + 1;
        v2f A = {rowmask * u1[k0], rowmask * u1[k1]};
        v2f B = {Wlp[k0 * 128 + col], Wlp[k1 * 128 + col]};
        c = __builtin_amdgcn_wmma_f32_16x16x4_f32(false, A, false, B,
                                                  (short)0, c, false, false);
    }
    // Pass 2: u2 @ (Wr_pos + Wr_neg)
    for (int kb = 0; kb < 128; kb += 4) {
        const int k0 = kb + 2 * half, k1 = k0 + 1;
        v2f A = {rowmask * u2[k0], rowmask * u2[k1]};
        v2f B = {Wrp[k0 * 128 + col] + Wrn[k0 * 128 + col],
                 Wrp[k1 * 128 + col] + Wrn[k1 * 128 + col]};
        c = __builtin_amdgcn_wmma_f32_16x16x4_f32(false, A, false, B,
                                                  (short)0, c, false, false);
    }
    // Pass 3: u3 @ Wl_neg
    for (int kb = 0; kb < 128; kb += 4) {
        const int k0 = kb + 2 * half, k1 = k0 + 1;
        v2f A = {rowmask * u3[k0], rowmask * u3[k1]};
        v2f B = {Wln[k0 * 128 + col], Wln[k1 * 128 + col]};
        c = __builtin_amdgcn_wmma_f32_16x16x4_f32(false, A, false, B,
                                                  (short)0, c, false, false);
    }

    if (lane < 16) out[col] = c[0] + blp[col] + bln[col];
}

// ---------------------------------------------------------------------------
// Launcher
// ---------------------------------------------------------------------------
extern "C" void kernel_launch(void* const* d_in, const int* in_sizes, int n_in,
                              void* d_out, int out_size, void* d_ws, size_t ws_size,
                              hipStream_t stream) {
    const float* x   = (const float*)d_in[0];
    const int*   eip = (const int*)d_in[1];   // [2,E] pos: src=[0,E), dst=[E,2E)
    const int*   ein = (const int*)d_in[2];   // [2,E] neg
    const float* Wlp = (const float*)d_in[3];
    const float* blp = (const float*)d_in[4];
    const float* Wrp = (const float*)d_in[5];
    const float* Wln = (const float*)d_in[6];
    const float* bln = (const float*)d_in[7];
    const float* Wrn = (const float*)d_in[8];

    const int F  = 128;
    const int N  = in_sizes[0] / F;
    const int Ep = in_sizes[1] / 2;
    const int En = in_sizes[2] / 2;

    // Workspace layout (floats): [cnt_pos N][cnt_neg N][a_pos N][a_neg N][vecs 384]
    float*    ws      = (float*)d_ws;
    unsigned* cnt_pos = (unsigned*)(ws);
    unsigned* cnt_neg = (unsigned*)(ws + N);
    float*    a_pos   = ws + 2 * (size_t)N;
    float*    a_neg   = ws + 3 * (size_t)N;
    float*    vecs    = ws + 4 * (size_t)N;

    const int zn = 4 * N + 384;
    zero_ws_kernel<<<(zn + 255) / 256, 256, 0, stream>>>(ws, zn);

    count_deg_kernel<<<(Ep + 255) / 256, 256, 0, stream>>>(eip + Ep, Ep, cnt_pos);
    count_deg_kernel<<<(En + 255) / 256, 256, 0, stream>>>(ein + En, En, cnt_neg);

    invert_cnt_kernel<<<(N + 255) / 256, 256, 0, stream>>>(cnt_pos, N);
    invert_cnt_kernel<<<(N + 255) / 256, 256, 0, stream>>>(cnt_neg, N);

    scatter_w_kernel<<<(Ep + 255) / 256, 256, 0, stream>>>(
        eip, eip + Ep, Ep, (const float*)cnt_pos, a_pos);
    scatter_w_kernel<<<(En + 255) / 256, 256, 0, stream>>>(
        ein, ein + En, En, (const float*)cnt_neg, a_neg);

    reduce_x_kernel<<<512, 128, 0, stream>>>(x, a_pos, a_neg, N, vecs);

    const float invN = 1.0f / (float)N;
    gemv_wmma_kernel<<<1, 256, 0, stream>>>(vecs, Wlp, Wrp, Wln, Wrn,
                                            blp, bln, (float*)d_out, invN);
}